// PointNetSetAbstraction_1726576857088
// MI455X (gfx1250) — compile-verified
//
#include <hip/hip_runtime.h>
#include <hip/hip_bf16.h>

typedef __attribute__((ext_vector_type(2))) float v2f;
typedef __attribute__((ext_vector_type(8))) float v8f;

#define NB      16
#define NPTS    4096
#define NPOINT  1024
#define NSAMP   32
#define RAD2    0.04f
constexpr size_t LTOT = (size_t)NB * NPOINT * NSAMP;   // 524288 columns

// ---------------------------------------------------------------------------
// 1) Farthest point sampling: one workgroup per batch, xyz + dist in LDS.
//    Writes new_xyz (B,3,NPOINT) directly into d_out[0 .. 49151].
// ---------------------------------------------------------------------------
__global__ __launch_bounds__(256) void fps_kernel(const float* __restrict__ xyz,
                                                  float* __restrict__ new_xyz)
{
    __shared__ float px[NPTS], py[NPTS], pz[NPTS], dist[NPTS];
    __shared__ float redv[256];
    __shared__ int   redi[256];
    __shared__ int   farLds;

    const int b   = blockIdx.x;
    const int tid = threadIdx.x;
    const float* xb = xyz + (size_t)b * 3 * NPTS;

    for (int i = tid; i < NPTS; i += 256) {
        px[i] = xb[i];
        py[i] = xb[NPTS + i];
        pz[i] = xb[2 * NPTS + i];
        dist[i] = 1e10f;
    }
    if (tid == 0) farLds = 0;
    __syncthreads();

    for (int it = 0; it < NPOINT; ++it) {
        const int far = farLds;
        const float cx = px[far], cy = py[far], cz = pz[far];
        if (tid == 0) {   // record sample `it` (old farthest, matches scan semantics)
            new_xyz[(b * 3 + 0) * NPOINT + it] = cx;
            new_xyz[(b * 3 + 1) * NPOINT + it] = cy;
            new_xyz[(b * 3 + 2) * NPOINT + it] = cz;
        }
        float bv = -1.0f; int bi = 0;
        const int p0 = tid * 16;
        #pragma unroll
        for (int p = p0; p < p0 + 16; ++p) {   // contiguous chunk -> tie-break = lowest index
            float dx = px[p] - cx, dy = py[p] - cy, dz = pz[p] - cz;
            float d  = dx * dx + dy * dy + dz * dz;
            float nd = fminf(dist[p], d);
            dist[p] = nd;
            if (nd > bv) { bv = nd; bi = p; }
        }
        redv[tid] = bv; redi[tid] = bi;
        __syncthreads();
        for (int s = 128; s > 0; s >>= 1) {
            if (tid < s) {
                float v2 = redv[tid + s]; int i2 = redi[tid + s];
                if (v2 > redv[tid] || (v2 == redv[tid] && i2 < redi[tid])) {
                    redv[tid] = v2; redi[tid] = i2;
                }
            }
            __syncthreads();
        }
        if (tid == 0) farLds = redi[0];
        __syncthreads();
    }
}

// ---------------------------------------------------------------------------
// 2) Ball query + grouping. One wave32 per centroid. Builds X0 (6 x LTOT),
//    column index l = (b*NPOINT + j)*NSAMP + k  (contiguous over k).
// ---------------------------------------------------------------------------
__global__ __launch_bounds__(256) void group_kernel(const float* __restrict__ xyz,
                                                    const float* __restrict__ pts,
                                                    const float* __restrict__ new_xyz,
                                                    float* __restrict__ X0)
{
    const int lane = threadIdx.x & 31;
    const int cid  = blockIdx.x * 8 + (threadIdx.x >> 5);   // centroid id
    const int b    = cid >> 10;
    const int j    = cid & 1023;

    const float* xb = xyz + (size_t)b * 3 * NPTS;
    const float cx = new_xyz[(b * 3 + 0) * NPOINT + j];
    const float cy = new_xyz[(b * 3 + 1) * NPOINT + j];
    const float cz = new_xyz[(b * 3 + 2) * NPOINT + j];

    int sel = 0, cnt = 0;
    for (int base = 0; base < NPTS; base += 32) {
        const int i = base + lane;
        float dx = xb[i] - cx, dy = xb[NPTS + i] - cy, dz = xb[2 * NPTS + i] - cz;
        bool inball = (dx * dx + dy * dy + dz * dz) <= RAD2;
        unsigned mask = (unsigned)__ballot(inball);
        int k = __popc(mask);
        int r = lane - cnt;               // my output slot relative to this chunk
        if (r >= 0 && r < k) {            // extract r-th set bit of mask
            unsigned m2 = mask;
            for (int t = 0; t < r; ++t) m2 &= (m2 - 1);
            sel = base + (__ffs(m2) - 1);
        }
        cnt += k;
        if (cnt >= NSAMP) break;
    }
    int first = __shfl(sel, 0, 32);
    if (lane >= cnt) sel = first;         // pad with first index (centroid guarantees cnt>=1)

    const size_t l = ((size_t)(b * NPOINT + j)) * NSAMP + lane;
    const float* pb = pts + (size_t)b * 3 * NPTS;
    X0[0 * LTOT + l] = xb[sel] - cx;
    X0[1 * LTOT + l] = xb[NPTS + sel] - cy;
    X0[2 * LTOT + l] = xb[2 * NPTS + sel] - cz;
    X0[3 * LTOT + l] = pb[sel];
    X0[4 * LTOT + l] = pb[NPTS + sel];
    X0[5 * LTOT + l] = pb[2 * NPTS + sel];
}

// ---------------------------------------------------------------------------
// 3) GEMM (Y = W*X + bias), fp32 WMMA 16x16x4, 256 columns per block,
//    double-buffered LDS X tiles (one barrier per subtile).
//      BN==true : previous layer's BN+ReLU fused into the reg->LDS stage,
//                 next subtile prefetched into registers during compute.
//      BN==false: pure copy -> CDNA5 async global->LDS DMA (ASYNCcnt),
//                 next subtile's DMA overlaps current subtile's WMMAs.
//    Fragment layouts per CDNA5 ISA 7.12.2:
//      A 16x4 : lanes0-15 K={k0,k0+1}, lanes16-31 K={k0+2,k0+3}, M = lane%16
//      B 4x16 : same K split across lane halves, N = lane%16
//      C 16x16: VGPR v -> M = v (lanes0-15) / v+8 (lanes16-31), N = lane%16
// ---------------------------------------------------------------------------
template <int CIN, int COUT, bool BN>
__global__ __launch_bounds__(256) void gemm_bn_kernel(const float* __restrict__ Xin,
                                                      const float* __restrict__ W,
                                                      const float* __restrict__ bias,
                                                      const float* __restrict__ statsIn,
                                                      float* __restrict__ Y)
{
    constexpr int KP   = (CIN + 3) & ~3;      // 8 for CIN=6, 64 otherwise
    constexpr int NT   = 64;                  // columns per subtile
    constexpr int NSUB = 4;                   // subtiles per block -> 256 columns
    constexpr int NF4  = KP * (NT / 4);       // float4 elements per subtile
    constexpr int NREG = (NF4 + 255) / 256;   // per-thread staging regs (BN path)

    __shared__ float Wl[COUT][KP + 1];        // +1 pad: conflict-free A reads
    __shared__ float Xl[2][KP][NT + 1];       // +1 pad: conflict-free B reads
    __shared__ float Bl[COUT];

    const int tid  = threadIdx.x;
    const size_t col0 = (size_t)blockIdx.x * (NT * NSUB);

    for (int i = tid; i < COUT * KP; i += 256) {
        int m = i / KP, k = i % KP;
        Wl[m][k] = (k < CIN) ? W[m * CIN + k] : 0.0f;
    }
    if (tid < COUT) Bl[tid] = bias[tid];

    const int lane   = tid & 31;
    const int wave   = tid >> 5;
    const int lane16 = lane & 15;
    const int khalf  = (lane < 16) ? 0 : 2;
    const int mhalf  = (lane < 16) ? 0 : 8;
    constexpr int NMB    = COUT / 16;
    constexpr int NTILES = NMB * (NT / 16);

    int buf = 0;

    float4 regs[NREG];
    if constexpr (BN) {
        // stage subtile 0 into registers
        #pragma unroll
        for (int t = 0; t < NREG; ++t) {
            int f = tid + t * 256;
            float4 v = {0.0f, 0.0f, 0.0f, 0.0f};
            if (f < NF4) {
                int k = f / (NT / 4), n4 = f % (NT / 4);
                if (k < CIN)
                    v = *(const float4*)&Xin[(size_t)k * LTOT + col0 + n4 * 4];
            }
            regs[t] = v;
        }
    } else {
        // zero the K-pad rows once (async DMA only touches rows k < CIN)
        for (int i = tid; i < 2 * (KP - CIN) * NT; i += 256) {
            int bb = i / ((KP - CIN) * NT);
            int r  = i % ((KP - CIN) * NT);
            Xl[bb][CIN + r / NT][r % NT] = 0.0f;
        }
        // launch async DMA for subtile 0 (one b128 per thread, threads < NF4)
        if (tid < NF4) {
            int k = tid / (NT / 4), n4 = tid % (NT / 4);
            if (k < CIN) {
                unsigned lds_off = (unsigned)(size_t)&Xl[0][k][n4 * 4];
                const float4* ga = (const float4*)&Xin[(size_t)k * LTOT + col0 + n4 * 4];
                asm volatile("global_load_async_to_lds_b128 %0, %1, off"
                             :: "v"(lds_off), "v"(ga) : "memory");
            }
        }
    }

    for (int s = 0; s < NSUB; ++s) {
        const size_t n0 = col0 + (size_t)s * NT;

        if constexpr (BN) {
            // write staged regs (BN + ReLU fused) into current LDS buffer
            #pragma unroll
            for (int t = 0; t < NREG; ++t) {
                int f = tid + t * 256;
                if (f < NF4) {
                    int k = f / (NT / 4), n4 = f % (NT / 4);
                    float4 v = regs[t];
                    if (k < CIN) {
                        float sc = statsIn[2 * k], sh = statsIn[2 * k + 1];
                        v.x = fmaxf(v.x * sc + sh, 0.0f);
                        v.y = fmaxf(v.y * sc + sh, 0.0f);
                        v.z = fmaxf(v.z * sc + sh, 0.0f);
                        v.w = fmaxf(v.w * sc + sh, 0.0f);
                    }
                    *(float4*)&Xl[buf][k][n4 * 4] = v;
                }
            }
            __syncthreads();                  // buf ready; compute(s-1) done everywhere
            if (s + 1 < NSUB) {               // prefetch next subtile during compute
                const size_t nc = col0 + (size_t)(s + 1) * NT;
                #pragma unroll
                for (int t = 0; t < NREG; ++t) {
                    int f = tid + t * 256;
                    float4 v = {0.0f, 0.0f, 0.0f, 0.0f};
                    if (f < NF4) {
                        int k = f / (NT / 4), n4 = f % (NT / 4);
                        if (k < CIN)
                            v = *(const float4*)&Xin[(size_t)k * LTOT + nc + n4 * 4];
                    }
                    regs[t] = v;
                }
            }
        } else {
            asm volatile("s_wait_asynccnt 0" ::: "memory");  // subtile s landed
            __syncthreads();                  // visible to all; compute(s-1) done
            if (s + 1 < NSUB) {               // DMA next subtile into other buffer
                if (tid < NF4) {
                    int k = tid / (NT / 4), n4 = tid % (NT / 4);
                    if (k < CIN) {
                        unsigned lds_off = (unsigned)(size_t)&Xl[buf ^ 1][k][n4 * 4];
                        const float4* ga = (const float4*)
                            &Xin[(size_t)k * LTOT + col0 + (size_t)(s + 1) * NT + n4 * 4];
                        asm volatile("global_load_async_to_lds_b128 %0, %1, off"
                                     :: "v"(lds_off), "v"(ga) : "memory");
                    }
                }
            }
        }

        // -------- compute: WMMA over this 64-column subtile --------
        for (int t = wave; t < NTILES; t += 8) {
            const int mb = (t % NMB) * 16;
            const int nb = (t / NMB) * 16;
            v8f acc = {};
            #pragma unroll
            for (int k0 = 0; k0 < KP; k0 += 4) {
                v2f a, bf;
                a.x  = Wl[mb + lane16][k0 + khalf];
                a.y  = Wl[mb + lane16][k0 + khalf + 1];
                bf.x = Xl[buf][k0 + khalf][nb + lane16];
                bf.y = Xl[buf][k0 + khalf + 1][nb + lane16];
                acc = __builtin_amdgcn_wmma_f32_16x16x4_f32(false, a, false, bf,
                                                            (short)0, acc, false, false);
            }
            #pragma unroll
            for (int v = 0; v < 8; ++v) {
                const int m = mb + mhalf + v;
                Y[(size_t)m * LTOT + n0 + nb + lane16] = acc[v] + Bl[m];
            }
        }
        buf ^= 1;
    }
}

// ---------------------------------------------------------------------------
// 4) Per-channel BN statistics -> (scale, shift). One block per channel,
//    fixed-partition tree reduction (deterministic).
// ---------------------------------------------------------------------------
__global__ __launch_bounds__(256) void reduce_bn_kernel(const float* __restrict__ Y,
                                                        const float* __restrict__ gamma,
                                                        const float* __restrict__ beta,
                                                        float* __restrict__ stats)
{
    __shared__ float s1[256], s2[256];
    const int c = blockIdx.x, tid = threadIdx.x;
    const float4* row = (const float4*)(Y + (size_t)c * LTOT);
    float s = 0.0f, q = 0.0f;
    for (int i = tid; i < (int)(LTOT / 4); i += 256) {
        float4 v = row[i];
        s += v.x + v.y + v.z + v.w;
        q += v.x * v.x + v.y * v.y + v.z * v.z + v.w * v.w;
    }
    s1[tid] = s; s2[tid] = q;
    __syncthreads();
    for (int st = 128; st > 0; st >>= 1) {
        if (tid < st) { s1[tid] += s1[tid + st]; s2[tid] += s2[tid + st]; }
        __syncthreads();
    }
    if (tid == 0) {
        float mean = s1[0] / (float)LTOT;
        float var  = s2[0] / (float)LTOT - mean * mean;
        float rsig = rsqrtf(var + 1e-5f);
        float scale = gamma[c] * rsig;
        stats[2 * c]     = scale;
        stats[2 * c + 1] = beta[c] - mean * scale;
    }
}

// ---------------------------------------------------------------------------
// 5) Final BN+ReLU + max over the 32 samples -> new_feats (B,128,NPOINT).
// ---------------------------------------------------------------------------
__global__ __launch_bounds__(256) void maxpool_kernel(const float* __restrict__ Y2,
                                                      const float* __restrict__ stats,
                                                      float* __restrict__ out)
{
    const int gid = blockIdx.x * 256 + threadIdx.x;   // B*128*NPOINT threads
    const int j = gid & 1023;
    const int c = (gid >> 10) & 127;
    const int b = gid >> 17;
    const float scale = stats[2 * c], shift = stats[2 * c + 1];
    const float4* p = (const float4*)(Y2 + (size_t)c * LTOT +
                                      ((size_t)(b * NPOINT + j)) * NSAMP);
    float m = 0.0f;   // ReLU outputs are >= 0, so 0 is the identity for this max
    #pragma unroll
    for (int t = 0; t < 8; ++t) {
        float4 v = p[t];
        m = fmaxf(m, fmaxf(fmaxf(v.x * scale + shift, v.y * scale + shift),
                           fmaxf(v.z * scale + shift, v.w * scale + shift)));
    }
    out[((size_t)(b * 128 + c) << 10) + j] = m;
}

// ---------------------------------------------------------------------------
// Host launcher. Inputs: xyz, points, then (w,b,gamma,beta) x 3 layers.
// d_out: new_xyz (16*3*1024 floats) ++ new_feats (16*128*1024 floats).
// ---------------------------------------------------------------------------
extern "C" void kernel_launch(void* const* d_in, const int* in_sizes, int n_in,
                              void* d_out, int out_size, void* d_ws, size_t ws_size,
                              hipStream_t stream)
{
    const float* xyz    = (const float*)d_in[0];
    const float* points = (const float*)d_in[1];
    const float* w0 = (const float*)d_in[2];
    const float* b0 = (const float*)d_in[3];
    const float* g0 = (const float*)d_in[4];
    const float* be0 = (const float*)d_in[5];
    const float* w1 = (const float*)d_in[6];
    const float* b1 = (const float*)d_in[7];
    const float* g1 = (const float*)d_in[8];
    const float* be1 = (const float*)d_in[9];
    const float* w2 = (const float*)d_in[10];
    const float* b2 = (const float*)d_in[11];
    const float* g2 = (const float*)d_in[12];
    const float* be2 = (const float*)d_in[13];

    float* out      = (float*)d_out;
    float* new_xyz  = out;                         // (B,3,NPOINT)
    float* new_feat = out + (size_t)NB * 3 * NPOINT;

    float* ws = (float*)d_ws;
    float* X0 = ws;                                // 6  * L
    float* Y0 = X0 + 6 * LTOT;                     // 64 * L
    float* Y1 = Y0 + 64 * LTOT;                    // 64 * L
    float* Y2 = Y1 + 64 * LTOT;                    // 128 * L
    float* st0 = Y2 + 128 * LTOT;                  // 128 floats
    float* st1 = st0 + 128;
    float* st2 = st1 + 128;

    fps_kernel<<<NB, 256, 0, stream>>>(xyz, new_xyz);
    group_kernel<<<(NB * NPOINT) / 8, 256, 0, stream>>>(xyz, points, new_xyz, X0);

    const int gblk = (int)(LTOT / 256);            // 2048 blocks, 256 columns each
    gemm_bn_kernel<6, 64, false><<<gblk, 256, 0, stream>>>(X0, w0, b0, nullptr, Y0);
    reduce_bn_kernel<<<64, 256, 0, stream>>>(Y0, g0, be0, st0);
    gemm_bn_kernel<64, 64, true><<<gblk, 256, 0, stream>>>(Y0, w1, b1, st0, Y1);
    reduce_bn_kernel<<<64, 256, 0, stream>>>(Y1, g1, be1, st1);
    gemm_bn_kernel<64, 128, true><<<gblk, 256, 0, stream>>>(Y1, w2, b2, st1, Y2);
    reduce_bn_kernel<<<128, 256, 0, stream>>>(Y2, g2, be2, st2);

    maxpool_kernel<<<(NB * 128 * NPOINT) / 256, 256, 0, stream>>>(Y2, st2, new_feat);
}